// FasterCosSimConv2D_69346541961457
// MI455X (gfx1250) — compile-verified
//
#include <hip/hip_runtime.h>
#include <hip/hip_bf16.h>

// ---------------------------------------------------------------------------
// FasterCosSimConv2D on MI455X (gfx1250, wave32, WMMA)
//
// Shapes: x[16,112,112,128] f32 NHWC, W[3,3,128,256] HWIO, out[16,110,110,256]
// Implicit GEMM: M=193600, K=1152, N=256  (~114 GFLOP) -> compute-bound at
// fp32. Core: V_WMMA_F32_16X16X4_F32 on LDS tiles, double-buffered via
// GLOBAL_LOAD_ASYNC_TO_LDS_B128 (ASYNCcnt pipeline).
// ---------------------------------------------------------------------------

typedef __attribute__((ext_vector_type(2))) float v2f;
typedef __attribute__((ext_vector_type(8))) float v8f;
typedef __attribute__((ext_vector_type(4))) int   v4i;

// Problem constants
#define IN_N   16
#define IN_H   112
#define IN_W   112
#define IN_C   128
#define OC     256
#define OH     110
#define OW     110
#define M_TOT  (IN_N*OH*OW)      // 193600, divisible by 64
#define K_TOT  1152              // 3*3*128

// Workspace layout (floats)
#define S_OFF   0                          // sum of x^2 over channels: 16*112*112
#define S_LEN   (IN_N*IN_H*IN_W)           // 200704
#define XN_OFF  (S_OFF + S_LEN)            // x_norm: 193600
#define WN_OFF  (XN_OFF + M_TOT)           // w_norm: 256
#define E_OFF   (WN_OFF + OC)              // exponent p^2/100: 256

// LDS tile strides (floats), padded for bank-conflict-free WMMA fragment reads
#define A_STR 36    // 32 cols + pad; 36*4B rows keep 16B alignment, distinct banks
#define B_STR 144   // 128 cols + pad; rows r and r+2 land in disjoint bank halves

// --- CDNA5 async global->LDS copy (ASYNCcnt), with safe fallback ----------
#if __has_builtin(__builtin_amdgcn_global_load_async_to_lds_b128)
#define USE_ASYNC_LDS 1
typedef __attribute__((address_space(1))) v4i as1_v4i;  // global int4
typedef __attribute__((address_space(3))) v4i as3_v4i;  // LDS int4
#else
#define USE_ASYNC_LDS 0
#endif

__device__ __forceinline__ void copy_b128(const float* gsrc, float* ldst) {
#if USE_ASYNC_LDS
  __builtin_amdgcn_global_load_async_to_lds_b128((as1_v4i*)gsrc, (as3_v4i*)ldst,
                                                 0, 0);
#else
  *(float4*)ldst = *(const float4*)gsrc;
#endif
}

// ---------------------------------------------------------------------------
// Kernel 1: per-output-channel filter norm + exponent (trivial)
// ---------------------------------------------------------------------------
__global__ void prep_kernel(const float* __restrict__ Wt,
                            const float* __restrict__ p,
                            const float* __restrict__ q,
                            float* __restrict__ ws) {
  const int oc = threadIdx.x;        // one block of 256
  const float qv = q[0];
  const float qq = qv * qv * 0.1f;
  float s = 0.f;
  for (int i = 0; i < K_TOT; ++i) {
    const float w = Wt[(size_t)i * OC + oc];
    s = fmaf(w, w, s);
  }
  ws[WN_OFF + oc] = sqrtf(s) + qq;
  const float pv = p[oc];
  ws[E_OFF + oc] = pv * pv * 0.01f;
}

// ---------------------------------------------------------------------------
// Kernel 2: s[n,h,w] = sum_c x^2   (one wave32 per pixel, shuffle reduce)
// ---------------------------------------------------------------------------
__global__ __launch_bounds__(256) void sumsq_kernel(const float* __restrict__ x,
                                                    float* __restrict__ ws) {
  const int tid  = threadIdx.x;
  const int pix  = blockIdx.x * 8 + (tid >> 5);   // 8 pixels per 256-thread block
  const int lane = tid & 31;
  const float4 v = *(const float4*)(x + (size_t)pix * IN_C + lane * 4);
  float ss = v.x * v.x + v.y * v.y + v.z * v.z + v.w * v.w;
  #pragma unroll
  for (int off = 16; off > 0; off >>= 1) ss += __shfl_down(ss, off, 32);
  if (lane == 0) ws[S_OFF + pix] = ss;
}

// ---------------------------------------------------------------------------
// Kernel 3: x_norm[m] = sqrt(3x3 window sum of s) + q^2/10
// ---------------------------------------------------------------------------
__global__ void xnorm_kernel(const float* __restrict__ s,
                             const float* __restrict__ q,
                             float* __restrict__ xn) {
  const int m = blockIdx.x * 256 + threadIdx.x;
  if (m >= M_TOT) return;
  const int n  = m / (OH * OW);
  const int r  = m % (OH * OW);
  const int oh = r / OW;
  const int ow = r % OW;
  const float* sp = s + (size_t)n * (IN_H * IN_W);
  float acc = 0.f;
  #pragma unroll
  for (int kh = 0; kh < 3; ++kh)
    #pragma unroll
    for (int kw = 0; kw < 3; ++kw)
      acc += sp[(oh + kh) * IN_W + (ow + kw)];
  const float qv = q[0];
  xn[m] = sqrtf(acc) + qv * qv * 0.1f;
}

// ---------------------------------------------------------------------------
// Kernel 4: implicit-GEMM conv + fused cosine-sim epilogue.
// Workgroup: 256 threads = 8 wave32s. Tile: 64 M x 128 N.
// waveM = wid&3 -> 16-row M block; waveN = wid>>2 -> 64-wide N half; each wave
// holds 4 accumulators (16x16 f32). K staged in 32-wide LDS slices (each slice
// sits inside one (kh,kw) tap with contiguous channels -> coalesced B128s),
// double-buffered through the async global->LDS engine.
// ---------------------------------------------------------------------------
__global__ __launch_bounds__(256) void conv_wmma_kernel(
    const float* __restrict__ x, const float* __restrict__ Wt,
    const float* __restrict__ bias, const float* __restrict__ ws,
    float* __restrict__ out) {
  __shared__ __align__(16) float As[2][64 * A_STR];
  __shared__ __align__(16) float Bs[2][32 * B_STR];

  const int tid  = threadIdx.x;
  const int lane = tid & 31;
  const int wid  = tid >> 5;
  const int m0   = blockIdx.x * 64;    // 3025 blocks
  const int n0   = blockIdx.y * 128;   // 2 blocks

  // --- A-load geometry: thread copies row ar (of 64), 8 floats at col ac ---
  const int ar = tid >> 2;
  const int ac = (tid & 3) * 8;
  const int am   = m0 + ar;
  const int an   = am / (OH * OW);
  const int arem = am % (OH * OW);
  const int aoh  = arem / OW;
  const int aow  = arem % OW;
  const float* xrow = x + (((size_t)an * IN_H + aoh) * IN_W + aow) * IN_C;

  // --- B-load geometry: thread copies channel-row br (of 32), 16 floats at bc ---
  const int br = tid >> 3;
  const int bc = (tid & 7) * 16;

  v8f acc[4] = {};

  const int waveM = wid & 3;
  const int waveN = wid >> 2;
  const int mL    = waveM * 16 + (lane & 15); // A fragment row
  const int khalf = lane >> 4;                // 0: K{0,1}, 1: K{2,3}

  // Issue one K-slice (6 async B128 copies per thread) into buffer `buf`.
  auto issue_slice = [&](int kk, int buf) {
    const int tap = kk >> 2;                  // (kh,kw) tap 0..8
    const int c0  = (kk & 3) * 32;            // channel block within tap
    const int kh  = tap / 3;
    const int kw  = tap % 3;
    const float* asrc = xrow + ((size_t)kh * IN_W + kw) * IN_C + c0 + ac;
    float* adst = &As[buf][ar * A_STR + ac];
    copy_b128(asrc,     adst);
    copy_b128(asrc + 4, adst + 4);
    const float* bsrc = Wt + ((size_t)(tap * IN_C + c0 + br)) * OC + n0 + bc;
    float* bdst = &Bs[buf][br * B_STR + bc];
    #pragma unroll
    for (int i = 0; i < 4; ++i) copy_b128(bsrc + i * 4, bdst + i * 4);
  };

  issue_slice(0, 0);

  for (int kk = 0; kk < 36; ++kk) {           // 36 slices of K=32
    const int cur = kk & 1;
    if (kk + 1 < 36) {
      issue_slice(kk + 1, cur ^ 1);           // prefetch next slice
#if USE_ASYNC_LDS
      // Async loads retire in order: <=6 outstanding means slice kk's 6 copies
      // from this wave are in LDS; the 6 just-issued may still be in flight.
      asm volatile("s_wait_asynccnt 0x6" ::: "memory");
#endif
    } else {
#if USE_ASYNC_LDS
      asm volatile("s_wait_asynccnt 0x0" ::: "memory");
#endif
    }
    __syncthreads();                          // all waves' copies visible

    const float* Ac = As[cur];
    const float* Bc = Bs[cur];
    #pragma unroll
    for (int k4 = 0; k4 < 8; ++k4) {          // 8 WMMA K-steps of 4
      const int kb = k4 * 4 + 2 * khalf;
      const v2f a = *(const v2f*)&Ac[mL * A_STR + kb];
      #pragma unroll
      for (int ni = 0; ni < 4; ++ni) {
        const int nL = waveN * 64 + ni * 16 + (lane & 15);
        v2f b;
        b.x = Bc[kb * B_STR + nL];
        b.y = Bc[(kb + 1) * B_STR + nL];
        acc[ni] = __builtin_amdgcn_wmma_f32_16x16x4_f32(
            false, a, false, b, (short)0, acc[ni], false, false);
      }
    }
    __syncthreads();  // all waves done reading `cur` before it is re-filled
  }

  // ---------------- fused epilogue ----------------
  // C/D layout: VGPR j -> M = j (lanes 0-15) / j+8 (lanes 16-31), N = lane&15.
  float xnv[8];
  #pragma unroll
  for (int j = 0; j < 8; ++j)
    xnv[j] = ws[XN_OFF + m0 + waveM * 16 + j + khalf * 8];

  #pragma unroll
  for (int ni = 0; ni < 4; ++ni) {
    const int oc = n0 + waveN * 64 + ni * 16 + (lane & 15);
    const float wn = ws[WN_OFF + oc];
    const float ee = ws[E_OFF + oc];
    const float bb = bias[oc];
    #pragma unroll
    for (int j = 0; j < 8; ++j) {
      const int m = m0 + waveM * 16 + j + khalf * 8;
      const float feats = acc[ni][j] + bb;
      const float r = fabsf(feats) / (xnv[j] * wn) + 1e-12f;
      float y = __expf(ee * __logf(r));          // r > 0 always
      y = (feats > 0.f) ? y : ((feats < 0.f) ? -y : 0.f);
      // Output is written once, never re-read: keep it out of L2 (which
      // usefully holds x across taps / overlapping M-tiles instead).
      __builtin_nontemporal_store(y, &out[(size_t)m * OC + oc]);
    }
  }
}

// ---------------------------------------------------------------------------
extern "C" void kernel_launch(void* const* d_in, const int* in_sizes, int n_in,
                              void* d_out, int out_size, void* d_ws, size_t ws_size,
                              hipStream_t stream) {
  const float* x = (const float*)d_in[0];
  const float* W = (const float*)d_in[1];
  const float* b = (const float*)d_in[2];
  const float* p = (const float*)d_in[3];
  const float* q = (const float*)d_in[4];
  float* out = (float*)d_out;
  float* ws  = (float*)d_ws;

  prep_kernel<<<1, 256, 0, stream>>>(W, p, q, ws);
  sumsq_kernel<<<S_LEN / 8, 256, 0, stream>>>(x, ws);
  xnorm_kernel<<<(M_TOT + 255) / 256, 256, 0, stream>>>(ws + S_OFF, q, ws + XN_OFF);

  dim3 grid(M_TOT / 64, OC / 128);
  conv_wmma_kernel<<<grid, 256, 0, stream>>>(x, W, b, ws, out);
}